// SpGraphAttentionLayer_70342974374588
// MI455X (gfx1250) — compile-verified
//
#include <hip/hip_runtime.h>
#include <math.h>

typedef __attribute__((ext_vector_type(2))) float v2f;
typedef __attribute__((ext_vector_type(4))) float f4;
typedef __attribute__((ext_vector_type(8))) float v8f;

#define IN_F   256
#define OUT_F  64
#define ALPHA  0.2f
#define KB     32
#define LSTR   36   // LDS row stride (floats): 16B aligned, conflict-free for 16-row column reads

// ---------------- float atomic max (int/uint monotonic trick) ----------------
__device__ __forceinline__ void atomic_max_f32(float* addr, float val) {
    if (val >= 0.0f) {
        atomicMax((int*)addr, __float_as_int(val));
    } else {
        atomicMin((unsigned int*)addr, __float_as_uint(val));
    }
}

// ---------------- init: zero output accumulator, -inf segment max, 0 denom ----------------
__global__ void k_init(float* __restrict__ out, float* __restrict__ m,
                       float* __restrict__ dn, int n_nodes) {
    int i = blockIdx.x * blockDim.x + threadIdx.x;
    if (i < n_nodes * OUT_F) out[i] = 0.0f;
    if (i < n_nodes) { m[i] = -__builtin_inff(); dn[i] = 0.0f; }
}

// ---------------- h = x @ W^T + b via V_WMMA_F32_16X16X4_F32 ----------------
// Block = 256 threads = 8 waves. Wave w computes rows [blk*128 + w*16, +16) x all 64 cols.
__global__ __launch_bounds__(256) void k_gemm(const float* __restrict__ x,
                                              const float* __restrict__ W,
                                              const float* __restrict__ bias,
                                              float* __restrict__ h, int n_nodes) {
    __shared__ float lx[128 * LSTR];   // x tile: 128 rows x 32 k
    __shared__ float lw[64 * LSTR];    // W tile: 64 n x 32 k (B[k][n] = W[n][k])

    const int tid      = threadIdx.x;
    const int wave     = tid >> 5;
    const int lane     = tid & 31;
    const int row_base = blockIdx.x * 128;
    const int lhalf    = lane & 15;    // m (A) / n (B) / n (C) index within tile
    const int kq       = lane >> 4;    // K-pair selector for A/B frags, M-half for C

    v8f acc[4] = {};                   // 4 n-tiles of 16 cols each

    for (int kb = 0; kb < IN_F; kb += KB) {
        // Cooperative load x tile: 128 rows x 32 floats, fully coalesced float4
        {
            const int r0 = tid >> 3;          // 0..31
            const int j  = tid & 7;           // float4 column
            #pragma unroll
            for (int i = 0; i < 4; ++i) {
                const int r    = r0 + 32 * i;
                const int grow = row_base + r;
                f4 v = (f4){0.f, 0.f, 0.f, 0.f};
                if (grow < n_nodes)
                    v = *(const f4*)(x + (size_t)grow * IN_F + kb + j * 4);
                *(f4*)(lx + r * LSTR + j * 4) = v;
            }
            // W tile: 64 rows x 32 floats
            #pragma unroll
            for (int i = 0; i < 2; ++i) {
                const int idx = tid + 256 * i;    // 0..511
                const int n   = idx >> 3;
                const int j2  = idx & 7;
                f4 v = *(const f4*)(W + (size_t)n * IN_F + kb + j2 * 4);
                *(f4*)(lw + n * LSTR + j2 * 4) = v;
            }
        }
        __syncthreads();

        const int arow = (wave << 4) + lhalf;     // local row in lx for this lane's A frag
        #pragma unroll
        for (int k = 0; k < KB; k += 4) {
            v2f a;
            a.x = lx[arow * LSTR + k + 2 * kq];
            a.y = lx[arow * LSTR + k + 2 * kq + 1];
            #pragma unroll
            for (int nt = 0; nt < 4; ++nt) {
                const int n = nt * 16 + lhalf;
                v2f b;
                b.x = lw[n * LSTR + k + 2 * kq];
                b.y = lw[n * LSTR + k + 2 * kq + 1];
                acc[nt] = __builtin_amdgcn_wmma_f32_16x16x4_f32(
                    /*neg_a=*/false, a, /*neg_b=*/false, b,
                    /*c_mod=*/(short)0, acc[nt],
                    /*reuse_a=*/false, /*reuse_b=*/false);
            }
        }
        __syncthreads();
    }

    // Epilogue: C layout — VGPR r: lanes 0-15 hold M=r, lanes 16-31 hold M=r+8; N = lane&15
    #pragma unroll
    for (int nt = 0; nt < 4; ++nt) {
        const int   col = nt * 16 + lhalf;
        const float bv  = bias[col];
        #pragma unroll
        for (int r = 0; r < 8; ++r) {
            const int grow = row_base + (wave << 4) + r + 8 * kq;
            if (grow < n_nodes)
                h[(size_t)grow * OUT_F + col] = acc[nt][r] + bv;
        }
    }
}

// ---------------- s1 = h @ a1, s2 = h @ a2 ----------------
__global__ void k_scores(const float* __restrict__ h, const float* __restrict__ a,
                         float* __restrict__ s1, float* __restrict__ s2, int n_nodes) {
    int i = blockIdx.x * blockDim.x + threadIdx.x;
    if (i >= n_nodes) return;
    const float* hr = h + (size_t)i * OUT_F;
    float d1 = 0.f, d2 = 0.f;
    #pragma unroll 8
    for (int f = 0; f < OUT_F; ++f) {
        const float v = hr[f];
        d1 += v * a[f];
        d2 += v * a[OUT_F + f];
    }
    s1[i] = d1;
    s2[i] = d2;
}

// ---------------- edge pass 1: e = leakyrelu(s1[src]+s2[dst]); segment max over dst ----------------
__global__ void k_edge1(const int* __restrict__ adj, const float* __restrict__ s1,
                        const float* __restrict__ s2, float* __restrict__ eb,
                        float* __restrict__ m, long long E) {
    long long t = blockIdx.x * (long long)blockDim.x + threadIdx.x;
    if (t >= E) return;
    const int src = adj[t];
    const int dst = adj[E + t];
    float e = s1[src] + s2[dst];
    e = e > 0.f ? e : ALPHA * e;
    eb[t] = e;
    atomic_max_f32(m + dst, e);
}

// ---------------- edge pass 2: ex = exp(e - m[dst]); segment sum denom ----------------
__global__ void k_edge2(const int* __restrict__ adj, float* __restrict__ eb,
                        const float* __restrict__ m, float* __restrict__ dn, long long E) {
    long long t = blockIdx.x * (long long)blockDim.x + threadIdx.x;
    if (t >= E) return;
    const int dst = adj[E + t];
    const float ex = expf(eb[t] - m[dst]);
    eb[t] = ex;
    atomicAdd(dn + dst, ex);
}

// ---------------- edge pass 3: out[src] += (ex/denom[dst]) * h[dst]  (wave per edge) ----------------
__global__ __launch_bounds__(256) void k_scatter(const int* __restrict__ adj,
                                                 const float* __restrict__ h,
                                                 const float* __restrict__ eb,
                                                 const float* __restrict__ dn,
                                                 float* __restrict__ out, long long E) {
    const long long e = (long long)blockIdx.x * 8 + (threadIdx.x >> 5);
    if (e >= E) return;
    const int lane = threadIdx.x & 31;
    const int src  = adj[e];
    const int dst  = adj[E + e];
    const float att = eb[e] / dn[dst];
    const float v0 = h[(size_t)dst * OUT_F + lane];
    const float v1 = h[(size_t)dst * OUT_F + lane + 32];
    atomicAdd(out + (size_t)src * OUT_F + lane,      att * v0);
    atomicAdd(out + (size_t)src * OUT_F + lane + 32, att * v1);
}

// ---------------- final ELU (in place on d_out) ----------------
__global__ void k_elu(float* __restrict__ out, long long total) {
    long long i = blockIdx.x * (long long)blockDim.x + threadIdx.x;
    if (i >= total) return;
    const float v = out[i];
    out[i] = v > 0.f ? v : (expf(v) - 1.f);
}

extern "C" void kernel_launch(void* const* d_in, const int* in_sizes, int n_in,
                              void* d_out, int out_size, void* d_ws, size_t ws_size,
                              hipStream_t stream) {
    const int*   adj = (const int*)d_in[0];
    const float* x   = (const float*)d_in[1];
    const float* W   = (const float*)d_in[2];
    const float* b   = (const float*)d_in[3];
    const float* a   = (const float*)d_in[4];
    float* out = (float*)d_out;

    const long long E       = in_sizes[0] / 2;
    const int       n_nodes = in_sizes[1] / IN_F;

    // Workspace layout (floats): h[N*64] | s1[N] | s2[N] | m[N] | denom[N] | eb[E]  (~40 MB)
    float* ws = (float*)d_ws;
    float* h  = ws;
    float* s1 = h  + (size_t)n_nodes * OUT_F;
    float* s2 = s1 + n_nodes;
    float* m  = s2 + n_nodes;
    float* dn = m  + n_nodes;
    float* eb = dn + n_nodes;

    const int tpb = 256;
    const long long tot_out = (long long)n_nodes * OUT_F;

    k_init<<<(int)((tot_out + tpb - 1) / tpb), tpb, 0, stream>>>(out, m, dn, n_nodes);

    const int mtiles  = (n_nodes + 15) / 16;
    const int gblocks = (mtiles + 7) / 8;
    k_gemm<<<gblocks, 256, 0, stream>>>(x, W, b, h, n_nodes);

    k_scores<<<(n_nodes + tpb - 1) / tpb, tpb, 0, stream>>>(h, a, s1, s2, n_nodes);

    k_edge1<<<(int)((E + tpb - 1) / tpb), tpb, 0, stream>>>(adj, s1, s2, eb, m, E);
    k_edge2<<<(int)((E + tpb - 1) / tpb), tpb, 0, stream>>>(adj, eb, m, dn, E);
    k_scatter<<<(int)((E + 7) / 8), 256, 0, stream>>>(adj, h, eb, dn, out, E);

    k_elu<<<(int)((tot_out + tpb - 1) / tpb), tpb, 0, stream>>>(out, tot_out);
}